// SpatialTemporalModel_12670153523505
// MI455X (gfx1250) — compile-verified
//
#include <hip/hip_runtime.h>
#include <hip/hip_bf16.h>

// ---------------------------------------------------------------------------
// GMAN-style spatio-temporal transformer forward, fp32, CDNA5 (gfx1250).
// All dense GEMMs use V_WMMA_F32_16X16X4_F32 (exact fp32 tensor-core path).
// Each wave computes a 64x16 output strip (4 accumulators) so every B
// fragment feeds 4 WMMAs; all loads are unconditional (clamped addresses,
// cndmask zeroing) so no EXEC-mask branching around the WMMA pipeline.
// ---------------------------------------------------------------------------

typedef __attribute__((ext_vector_type(2))) float v2f;
typedef __attribute__((ext_vector_type(8))) float v8f;

#define BB   64
#define PP   12
#define QQ   12
#define NN   325
#define DD   64
#define RR   (BB*PP*NN)      /* 249600 rows, divisible by 16 */
#define BT   (BB*PP)         /* 768 (b,t) slices */
#define NEGV (-9.0e15f)

__device__ __forceinline__ float gelu_f(float x) {
    // exact gelu: 0.5*x*(1+erf(x/sqrt(2)))
    return 0.5f * x * (1.0f + erff(x * 0.70710678118654752f));
}

struct SrcPtrs { const float* p[7]; };

// ---------------------------------------------------------------------------
// Fused linear: out[M,64] = act( concat(srcs)[M, NSRC*64] @ W + bias (+addend) )
// Block = 128 threads = 4 waves; wave w owns columns [16w,16w+16); block owns
// 64 rows (4 row-tiles of 16). act: 0=none, 1=gelu, 2=sigmoid
// ---------------------------------------------------------------------------
template<int NSRC>
__global__ __launch_bounds__(128)
void fused_linear_kernel(SrcPtrs srcs,
                         const float* __restrict__ W,
                         const float* __restrict__ bias,
                         const float* __restrict__ addend,
                         float* __restrict__ out, int M, int act)
{
    int lane   = threadIdx.x & 31;
    int wave   = threadIdx.x >> 5;
    int l16    = lane & 15;
    int half   = lane >> 4;              // 0 or 1
    int rowB   = blockIdx.x << 6;        // 64 rows per block
    int col    = (wave << 4) + l16;

    // clamped (always in-bounds) A-row indices for the 4 row tiles
    int mr[4];
    #pragma unroll
    for (int t = 0; t < 4; ++t) {
        int r = rowB + (t << 4) + l16;
        mr[t] = (r < M) ? r : (M - 1);
    }

    v8f acc[4] = {{}, {}, {}, {}};
    #pragma unroll
    for (int s = 0; s < NSRC; ++s) {
        const float* A  = srcs.p[s];
        const float* Wb = W + (s << 6) * 64;
        #pragma unroll
        for (int kk = 0; kk < 64; kk += 4) {
            int k = kk + (half << 1);
            v2f b;
            b.x = Wb[k * 64 + col];
            b.y = Wb[(k + 1) * 64 + col];
            v2f a0 = *(const v2f*)(A + mr[0] * 64 + k);
            v2f a1 = *(const v2f*)(A + mr[1] * 64 + k);
            v2f a2 = *(const v2f*)(A + mr[2] * 64 + k);
            v2f a3 = *(const v2f*)(A + mr[3] * 64 + k);
            acc[0] = __builtin_amdgcn_wmma_f32_16x16x4_f32(false, a0, false, b, (short)0, acc[0], false, false);
            acc[1] = __builtin_amdgcn_wmma_f32_16x16x4_f32(false, a1, false, b, (short)0, acc[1], false, false);
            acc[2] = __builtin_amdgcn_wmma_f32_16x16x4_f32(false, a2, false, b, (short)0, acc[2], false, false);
            acc[3] = __builtin_amdgcn_wmma_f32_16x16x4_f32(false, a3, false, b, (short)0, acc[3], false, false);
        }
    }
    float bv = bias[col];
    #pragma unroll
    for (int t = 0; t < 4; ++t) {
        #pragma unroll
        for (int r = 0; r < 8; ++r) {
            int row = rowB + (t << 4) + (half << 3) + r;
            if (row < M) {
                float v = acc[t][r] + bv;
                if (addend) v += addend[row * 64 + col];
                if (act == 1)      v = gelu_f(v);
                else if (act == 2) v = 1.0f / (1.0f + expf(-v));
                out[row * 64 + col] = v;
            }
        }
    }
}

// ---------------------------------------------------------------------------
// Graph GEMM: per (b,t) slice, out_bt[325,64] = op(A)[325,325] @ H_bt[325,64]
// TRANSA=1 -> op(A)[m][k] = A[k][m]  (gcn einsum 'btvd,vw->btwd')
// Block owns 64 rows x 64 cols; grid = (ceil(325/64)=6, 768).
// K tail (325 % 4 != 0) handled by clamped loads + A-side zeroing.
// ---------------------------------------------------------------------------
template<int TRANSA>
__global__ __launch_bounds__(128)
void graph_gemm_kernel(const float* __restrict__ Amat,
                       const float* __restrict__ H,
                       float* __restrict__ out, int do_gelu)
{
    int lane = threadIdx.x & 31;
    int wave = threadIdx.x >> 5;
    int l16  = lane & 15;
    int half = lane >> 4;
    int bt   = blockIdx.y;
    int rowB = blockIdx.x << 6;
    int col  = (wave << 4) + l16;
    const float* Hb = H + bt * (NN * 64);

    int mr[4];
    #pragma unroll
    for (int t = 0; t < 4; ++t) {
        int r = rowB + (t << 4) + l16;
        mr[t] = (r < NN) ? r : (NN - 1);
    }

    v8f acc[4] = {{}, {}, {}, {}};
    for (int kk = 0; kk < NN; kk += 4) {
        int k  = kk + (half << 1);
        int k0 = (k     < NN) ? k     : (NN - 1);
        int k1 = (k + 1 < NN) ? k + 1 : (NN - 1);
        v2f b;
        b.x = Hb[k0 * 64 + col];
        b.y = Hb[k1 * 64 + col];
        v2f a[4];
        #pragma unroll
        for (int t = 0; t < 4; ++t) {
            float ax, ay;
            if (TRANSA) { ax = Amat[k0 * NN + mr[t]]; ay = Amat[k1 * NN + mr[t]]; }
            else        { ax = Amat[mr[t] * NN + k0]; ay = Amat[mr[t] * NN + k1]; }
            a[t].x = (k     < NN) ? ax : 0.0f;   // zero tail contributions
            a[t].y = (k + 1 < NN) ? ay : 0.0f;
        }
        acc[0] = __builtin_amdgcn_wmma_f32_16x16x4_f32(false, a[0], false, b, (short)0, acc[0], false, false);
        acc[1] = __builtin_amdgcn_wmma_f32_16x16x4_f32(false, a[1], false, b, (short)0, acc[1], false, false);
        acc[2] = __builtin_amdgcn_wmma_f32_16x16x4_f32(false, a[2], false, b, (short)0, acc[2], false, false);
        acc[3] = __builtin_amdgcn_wmma_f32_16x16x4_f32(false, a[3], false, b, (short)0, acc[3], false, false);
    }
    #pragma unroll
    for (int t = 0; t < 4; ++t) {
        #pragma unroll
        for (int r = 0; r < 8; ++r) {
            int row = rowB + (t << 4) + (half << 3) + r;
            if (row < NN) {
                float v = acc[t][r];
                if (do_gelu) v = gelu_f(v);
                out[bt * (NN * 64) + row * 64 + col] = v;
            }
        }
    }
}

// nv1[325,10] @ nv2[10,325]
__global__ void nvprod_kernel(const float* __restrict__ nv1,
                              const float* __restrict__ nv2,
                              float* __restrict__ out)
{
    int idx = blockIdx.x * blockDim.x + threadIdx.x;
    if (idx >= NN * NN) return;
    int v = idx / NN, w = idx % NN;
    float s = 0.0f;
    #pragma unroll
    for (int i = 0; i < 10; ++i) s += nv1[v * 10 + i] * nv2[i * NN + w];
    out[idx] = s;
}

// in-place row softmax over 325 cols; optional adjacency mask / relu pre-op
__global__ void row_softmax_kernel(float* __restrict__ mat,
                                   const unsigned char* __restrict__ mask,
                                   int relu)
{
    __shared__ float red[128];
    int row = blockIdx.x, tid = threadIdx.x;
    float* rp = mat + row * NN;
    const unsigned char* mp = mask ? mask + row * NN : nullptr;

    float mx = -3.0e38f;
    for (int c = tid; c < NN; c += 128) {
        float v = rp[c];
        if (relu) v = fmaxf(v, 0.0f);
        if (mp && !mp[c]) v = NEGV;
        mx = fmaxf(mx, v);
    }
    red[tid] = mx; __syncthreads();
    for (int s = 64; s > 0; s >>= 1) {
        if (tid < s) red[tid] = fmaxf(red[tid], red[tid + s]);
        __syncthreads();
    }
    mx = red[0]; __syncthreads();

    float sum = 0.0f;
    for (int c = tid; c < NN; c += 128) {
        float v = rp[c];
        if (relu) v = fmaxf(v, 0.0f);
        if (mp && !mp[c]) v = NEGV;
        v = expf(v - mx);
        rp[c] = v;
        sum += v;
    }
    red[tid] = sum; __syncthreads();
    for (int s = 64; s > 0; s >>= 1) {
        if (tid < s) red[tid] += red[tid + s];
        __syncthreads();
    }
    float inv = 1.0f / red[0];
    __syncthreads();
    for (int c = tid; c < NN; c += 128) rp[c] *= inv;
}

// MHA, heads=8, d=8, T<=12; one thread per (b,n,h,qt)
__global__ void mha_kernel(const float* __restrict__ q,
                           const float* __restrict__ k,
                           const float* __restrict__ v,
                           float* __restrict__ out, int Tq, int Tk, int causal)
{
    int idx = blockIdx.x * blockDim.x + threadIdx.x;
    int total = BB * NN * 8 * Tq;
    if (idx >= total) return;
    int qt = idx % Tq; int t1 = idx / Tq;
    int h  = t1 & 7;   int t2 = t1 >> 3;
    int n  = t2 % NN;  int b  = t2 / NN;

    const float* qp = q + ((b * Tq + qt) * NN + n) * 64 + h * 8;
    float qv[8];
    #pragma unroll
    for (int j = 0; j < 8; ++j) qv[j] = qp[j];

    float s[16];
    float mx = -3.0e38f;
    for (int kt = 0; kt < Tk; ++kt) {
        float val;
        if (causal && kt > qt) val = NEGV;
        else {
            const float* kp = k + ((b * Tk + kt) * NN + n) * 64 + h * 8;
            float d = 0.0f;
            #pragma unroll
            for (int j = 0; j < 8; ++j) d += qv[j] * kp[j];
            val = d * 0.35355339059327373f;   // 1/sqrt(8)
        }
        s[kt] = val; mx = fmaxf(mx, val);
    }
    float sum = 0.0f;
    for (int kt = 0; kt < Tk; ++kt) { s[kt] = expf(s[kt] - mx); sum += s[kt]; }
    float inv = 1.0f / sum;
    float o[8];
    #pragma unroll
    for (int j = 0; j < 8; ++j) o[j] = 0.0f;
    for (int kt = 0; kt < Tk; ++kt) {
        float a = s[kt] * inv;
        const float* vp = v + ((b * Tk + kt) * NN + n) * 64 + h * 8;
        #pragma unroll
        for (int j = 0; j < 8; ++j) o[j] += a * vp[j];
    }
    float* op = out + ((b * Tq + qt) * NN + n) * 64 + h * 8;
    #pragma unroll
    for (int j = 0; j < 8; ++j) op[j] = o[j];
}

// fc1: out[m][j] = gelu(X[m]*w[j] + b[j])
__global__ void fc1_kernel(const float* __restrict__ X, const float* __restrict__ w,
                           const float* __restrict__ b, float* __restrict__ out, int M)
{
    int idx = blockIdx.x * blockDim.x + threadIdx.x;
    if (idx >= M * 64) return;
    int m = idx >> 6, j = idx & 63;
    out[idx] = gelu_f(X[m] * w[j] + b[j]);
}

// one-hot TE @ W5 == row gather; te1[m][j] = gelu(W5[dow]+W5[7+tod]+b5)
__global__ void te_gather_kernel(const int* __restrict__ TE, const float* __restrict__ w5,
                                 const float* __restrict__ b5, float* __restrict__ out)
{
    int idx = blockIdx.x * blockDim.x + threadIdx.x;
    if (idx >= BB * 24 * 64) return;
    int j = idx & 63, m = idx >> 6;
    int dow = TE[m * 2 + 0];
    int tod = TE[m * 2 + 1];
    out[idx] = gelu_f(w5[dow * 64 + j] + w5[(7 + tod) * 64 + j] + b5[j]);
}

// STE[b,t,n,:] = se[n,:] + te2[b*24+t,:], split into STE_P (t<12) / STE_Q
__global__ void ste_kernel(const float* __restrict__ se, const float* __restrict__ te2,
                           float* __restrict__ steP, float* __restrict__ steQ)
{
    int idx = blockIdx.x * blockDim.x + threadIdx.x;
    if (idx >= BB * 24 * NN * 64) return;
    int j = idx & 63; int r = idx >> 6;
    int n = r % NN; int bt = r / NN;
    int t = bt % 24; int b = bt / 24;
    float val = se[n * 64 + j] + te2[bt * 64 + j];
    if (t < 12) steP[(((b * 12 + t) * NN) + n) * 64 + j] = val;
    else        steQ[(((b * 12 + (t - 12)) * NN) + n) * 64 + j] = val;
}

__global__ void mix_kernel(const float* __restrict__ z, const float* __restrict__ HS,
                           const float* __restrict__ HT, float* __restrict__ out, int n)
{
    int idx = blockIdx.x * blockDim.x + threadIdx.x;
    if (idx >= n) return;
    float zz = z[idx];
    out[idx] = zz * HS[idx] + (1.0f - zz) * HT[idx];
}

__global__ void add_kernel(const float* __restrict__ a, const float* __restrict__ b,
                           float* __restrict__ out, int n)
{
    int idx = blockIdx.x * blockDim.x + threadIdx.x;
    if (idx < n) out[idx] = a[idx] + b[idx];
}

// fc27: out[m] = g[m,:] . w27 + b27
__global__ void fc_out_kernel(const float* __restrict__ g, const float* __restrict__ w27,
                              const float* __restrict__ b27, float* __restrict__ out, int M)
{
    int m = blockIdx.x * blockDim.x + threadIdx.x;
    if (m >= M) return;
    float s = b27[0];
    #pragma unroll
    for (int j = 0; j < 64; ++j) s += g[m * 64 + j] * w27[j];
    out[m] = s;
}

// ---------------------------------------------------------------------------
// Host orchestration
// ---------------------------------------------------------------------------
extern "C" void kernel_launch(void* const* d_in, const int* in_sizes, int n_in,
                              void* d_out, int out_size, void* d_ws, size_t ws_size,
                              hipStream_t stream)
{
    auto F = [&](int i) { return (const float*)d_in[i]; };
    const float*         X    = F(0);
    const float*         SE   = F(1);
    const int*           TE   = (const int*)d_in[2];
    const unsigned char* adj0 = (const unsigned char*)d_in[3];
    const unsigned char* adj1 = (const unsigned char*)d_in[4];

    // Flattened param indices (depth-first, dict insertion order).
    // 5..8: fc1.{w,b}, fc2.{w,b}; 9..16: ste fc3..fc6; enc block @17, dec @55,
    // tr @93 (fc21..fc25), 103/104 fc26, 105/106 fc27, 107 nv1, 108 nv2.
    const int ENC = 17, DEC = 55, TR = 93;

    // workspace arena: 12 slots of [RR,64] f32 + small region
    const size_t U = (size_t)RR * 64;
    float* ws = (float*)d_ws;
    float* S[12];
    for (int i = 0; i < 12; ++i) S[i] = ws + (size_t)i * U;
    float* small = ws + 12 * U;
    float* adp   = small;           // 325*325
    float* amat  = adp  + NN * NN;  // 325*325
    float* se_t  = amat + NN * NN;  // 325*64
    float* se    = se_t + NN * 64;
    float* te1   = se   + NN * 64;  // 64*24*64
    float* te2   = te1  + BB * 24 * 64;
    float* steP  = S[10];
    float* steQ  = S[11];

    auto lin = [&](const float* s0, const float* s1, const float* s2, const float* s3,
                   const float* s4, const float* s5, const float* s6, int nsrc,
                   int wi, const float* addend, float* out, int M, int act) {
        SrcPtrs sp{};
        sp.p[0]=s0; sp.p[1]=s1; sp.p[2]=s2; sp.p[3]=s3; sp.p[4]=s4; sp.p[5]=s5; sp.p[6]=s6;
        dim3 g((M + 63) / 64);
        if (nsrc == 1)
            fused_linear_kernel<1><<<g, 128, 0, stream>>>(sp, F(wi), F(wi+1), addend, out, M, act);
        else if (nsrc == 2)
            fused_linear_kernel<2><<<g, 128, 0, stream>>>(sp, F(wi), F(wi+1), addend, out, M, act);
        else
            fused_linear_kernel<7><<<g, 128, 0, stream>>>(sp, F(wi), F(wi+1), addend, out, M, act);
    };
    auto lin1 = [&](const float* a, int wi, float* out, int M, int act,
                    const float* addend = nullptr) {
        lin(a, nullptr, nullptr, nullptr, nullptr, nullptr, nullptr, 1, wi, addend, out, M, act);
    };
    auto lin2 = [&](const float* a, const float* b, int wi, float* out, int act) {
        lin(a, b, nullptr, nullptr, nullptr, nullptr, nullptr, 2, wi, nullptr, out, RR, act);
    };
    auto ggemm = [&](const float* Amat, const float* H, float* out, int transA, int do_gelu) {
        dim3 g((NN + 63) / 64, BT);
        if (transA) graph_gemm_kernel<1><<<g, 128, 0, stream>>>(Amat, H, out, do_gelu);
        else        graph_gemm_kernel<0><<<g, 128, 0, stream>>>(Amat, H, out, do_gelu);
    };
    auto EW = [&](int n) { return dim3((n + 255) / 256); };

    // GAT: a = softmax(where(adj, nv1@nv2, NEG)); out = gelu(a @ gelu(lin([X,STE])))
    auto gat_f = [&](int gb, const float* Xin, const float* STEp,
                     const unsigned char* adj, float* out) {
        nvprod_kernel<<<EW(NN * NN), 256, 0, stream>>>(F(gb + 2), F(gb + 3), amat);
        row_softmax_kernel<<<NN, 128, 0, stream>>>(amat, adj, 0);
        lin2(Xin, STEp, gb, S[7], 1);
        ggemm(amat, S[7], out, 0, 1);
    };
    // GCN: out = adp^T @ gelu(lin([X,STE]))
    auto gcn_f = [&](int bb, const float* Xin, const float* STEp, float* out) {
        lin2(Xin, STEp, bb + 2, S[7], 1);
        ggemm(adp, S[7], out, 1, 0);
    };

    auto st_block = [&](const float* Xin, const float* STEp,
                        float* dest, int bb) {
        // spatial branch
        gat_f(bb + 4,  Xin,  STEp, adj0, S[1]);
        gat_f(bb + 8,  S[1], STEp, adj0, S[2]);
        gat_f(bb + 12, Xin,  STEp, adj1, S[3]);
        gat_f(bb + 16, S[3], STEp, adj1, S[4]);
        gcn_f(bb, Xin,  STEp, S[5]);
        gcn_f(bb, S[5], STEp, S[6]);
        lin(Xin, S[1], S[2], S[3], S[4], S[5], S[6], 7, bb + 0, nullptr, S[8], RR, 1); // HS
        // temporal branch (causal MHA)
        int ta = bb + 20;
        lin2(Xin, STEp, ta + 0, S[1], 1);  // q
        lin2(Xin, STEp, ta + 2, S[2], 1);  // k
        lin2(Xin, STEp, ta + 4, S[3], 1);  // v
        mha_kernel<<<EW(BB * NN * 8 * 12), 256, 0, stream>>>(S[1], S[2], S[3], S[4], 12, 12, 1);
        lin1(S[4], ta + 6, S[5], RR, 1);
        lin1(S[5], ta + 8, S[6], RR, 0);   // HT
        // gated fusion + residual
        int gf = bb + 30;
        lin1(S[8], gf + 0, S[7], RR, 0);                 // lin(HS,fc17)
        lin1(S[6], gf + 2, S[7], RR, 2, S[7]);           // z = sigmoid(+lin(HT,fc18))
        mix_kernel<<<EW(RR * 64), 256, 0, stream>>>(S[7], S[8], S[6], S[1], RR * 64);
        lin1(S[1], gf + 4, S[2], RR, 1);
        lin1(S[2], gf + 6, S[3], RR, 0);
        add_kernel<<<EW(RR * 64), 256, 0, stream>>>(Xin, S[3], dest, RR * 64);
    };

    // ---- global adaptive adjacency: softmax(relu(nv1@nv2), axis=1) ----
    nvprod_kernel<<<EW(NN * NN), 256, 0, stream>>>(F(107), F(108), adp);
    row_softmax_kernel<<<NN, 128, 0, stream>>>(adp, nullptr, 1);

    // ---- input embedding: Xe = lin(gelu(lin(X[...,None],fc1)),fc2) ----
    fc1_kernel<<<EW(RR * 64), 256, 0, stream>>>(X, F(5), F(6), S[1], RR);
    lin1(S[1], 7, S[0], RR, 0);                           // Xe -> S0

    // ---- STE embedding ----
    lin1(SE, 9, se_t, NN, 1);
    lin1(se_t, 11, se, NN, 0);
    te_gather_kernel<<<EW(BB * 24 * 64), 256, 0, stream>>>(TE, F(13), F(14), te1);
    lin1(te1, 15, te2, BB * 24, 0);
    ste_kernel<<<EW(BB * 24 * NN * 64), 256, 0, stream>>>(se, te2, steP, steQ);

    // ---- encoder ----
    st_block(S[0], steP, S[9], ENC);                      // H_enc -> S9

    // ---- transform attention ----
    lin1(steQ, TR + 0, S[1], RR, 1);                      // q
    lin1(steP, TR + 2, S[2], RR, 1);                      // k
    lin1(S[9], TR + 4, S[3], RR, 1);                      // v
    mha_kernel<<<EW(BB * NN * 8 * 12), 256, 0, stream>>>(S[1], S[2], S[3], S[4], 12, 12, 0);
    lin1(S[4], TR + 6, S[5], RR, 1);
    lin1(S[5], TR + 8, S[0], RR, 0);                      // Ht -> S0

    // ---- decoder ----
    st_block(S[0], steQ, S[9], DEC);                      // H_dec -> S9

    // ---- output head ----
    lin1(S[9], 103, S[1], RR, 1);
    fc_out_kernel<<<EW(RR), 256, 0, stream>>>(S[1], F(105), F(106), (float*)d_out, RR);

    (void)in_sizes; (void)n_in; (void)out_size; (void)ws_size;
}